// GlobalNormFlipFlop_14834817040685
// MI455X (gfx1250) — compile-verified
//
#include <hip/hip_runtime.h>
#include <math.h>
#include <stdint.h>

typedef __attribute__((ext_vector_type(2))) float v2f;
typedef __attribute__((ext_vector_type(4))) float v4f;
typedef __attribute__((ext_vector_type(8))) float v8f;

#define TSEQ   2048
#define NSEQ   256
#define KDIM   256   // INSIZE
#define CDIM   40    // 2*nbase*(nbase+1)

#define WAVES  8
#define MTILE  (16 * WAVES)   // 128 rows per block
#define KCHUNK 16
#define BPITCH 96             // words per K-pair row: 48 n-pairs; 96%64==32 ->
                              // half-waves hit disjoint bank ranges
#define XPITCH 20             // 16+4 floats: 80B rows (16B aligned), conflict-free

// ---------------------------------------------------------------------------
// Kernel 1: y = 5*tanh(x @ W^T + b), WMMA f32 16x16x4
// x: (M=T*N, 256) row major, W: (40, 256) row major, out: (M, 40)
// B staged in LDS as K-pairs: blds[k/2][n] = (B[k][n], B[k+1][n]) so each
// fragment is one aligned ds_load_b64 into a contiguous VGPR pair.
// x tiles staged per-wave via GLOBAL_LOAD_ASYNC_TO_LDS_B128 (ASYNCcnt).
// ---------------------------------------------------------------------------
__global__ __launch_bounds__(256) void gemm_tanh_kernel(
    const float* __restrict__ x, const float* __restrict__ W,
    const float* __restrict__ bias, float* __restrict__ out)
{
    __shared__ float blds[(KDIM / 2) * BPITCH];  // 128 pair-rows x 96 words
    __shared__ float xs[WAVES * 16 * XPITCH];    // per-wave 16 x 16 x staging

    const int tid  = threadIdx.x;
    const int lane = tid & 31;
    const int wv   = tid >> 5;
    const int m0   = blockIdx.x * MTILE + wv * 16;

    // B[k][n] = W[n][k], packed as K-pairs: word = (k/2)*96 + 2n + (k&1)
    for (int i = tid; i < CDIM * KDIM; i += 256) {
        int n = i >> 8;        // 0..39
        int k = i & 255;
        blds[(k >> 1) * BPITCH + (n << 1) + (k & 1)] = W[i];
    }
    // zero-pad columns 40..47
    for (int i = tid; i < 8 * KDIM; i += 256) {
        int k = i >> 3;
        int n = CDIM + (i & 7);
        blds[(k >> 1) * BPITCH + (n << 1) + (k & 1)] = 0.0f;
    }
    __syncthreads();

    v8f acc0 = {}, acc1 = {}, acc2 = {};

    const int half = lane >> 4;     // 0: K=0,1 ; 1: K=2,3
    const int l15  = lane & 15;
    const int koff = half << 1;

    float* xw = &xs[wv * 16 * XPITCH];

    for (int kc = 0; kc < KDIM; kc += KCHUNK) {
        // Async-stage x[m0..m0+15][kc..kc+15] -> LDS, 16B per lane per op
        #pragma unroll
        for (int it = 0; it < 2; ++it) {
            int idx = it * 32 + lane;        // 0..63 float4 units
            int row = idx >> 2;              // 0..15
            int c4  = (idx & 3) << 2;        // 0,4,8,12
            uint64_t ga = (uint64_t)(uintptr_t)(x + (size_t)(m0 + row) * KDIM + kc + c4);
            uint32_t la = (uint32_t)(uintptr_t)(&xw[row * XPITCH + c4]);
            asm volatile("global_load_async_to_lds_b128 %0, %1, off"
                         :: "v"(la), "v"(ga) : "memory");
        }
        asm volatile("s_wait_asynccnt 0" ::: "memory");

        #pragma unroll
        for (int k4 = 0; k4 < KCHUNK; k4 += 4) {
            const int kk = k4 + koff;                  // local A col, this half
            const int kg2 = (kc + k4) >> 1;            // global K-pair index

            // A fragment (16x4): contiguous aligned pair -> ds_load_b64
            v2f a = *(const v2f*)&xw[l15 * XPITCH + kk];

            // B fragments (4x16): one aligned b64 per n-tile; half-wave adds
            // one pair-row (+96 words). n-tile offsets +32/+64 words.
            const int bb = (kg2 + half) * BPITCH + (l15 << 1);
            v2f b0 = *(const v2f*)&blds[bb];
            v2f b1 = *(const v2f*)&blds[bb + 32];
            v2f b2 = *(const v2f*)&blds[bb + 64];

            // 8 args: (neg_a, A, neg_b, B, c_mod, C, reuse_a, reuse_b)
            acc0 = __builtin_amdgcn_wmma_f32_16x16x4_f32(
                       false, a, false, b0, (short)0, acc0, false, false);
            acc1 = __builtin_amdgcn_wmma_f32_16x16x4_f32(
                       false, a, false, b1, (short)0, acc1, false, false);
            acc2 = __builtin_amdgcn_wmma_f32_16x16x4_f32(
                       false, a, false, b2, (short)0, acc2, false, false);
        }
    }

    // Epilogue: C layout -> VGPR r: lanes0-15 row r, lanes16-31 row r+8
    const int rowoff = half * 8;
    const float bia0 = bias[l15];
    const float bia1 = bias[16 + l15];
    const int   n2   = 32 + l15;
    const float bia2 = (n2 < CDIM) ? bias[n2] : 0.0f;

    #pragma unroll
    for (int r = 0; r < 8; ++r) {
        size_t grow = (size_t)(m0 + r + rowoff) * CDIM;
        out[grow + l15]      = 5.0f * tanhf(acc0[r] + bia0);
        out[grow + 16 + l15] = 5.0f * tanhf(acc1[r] + bia1);
        if (n2 < CDIM)
            out[grow + n2]   = 5.0f * tanhf(acc2[r] + bia2);
    }
}

// ---------------------------------------------------------------------------
// Kernel 2: sequential flip-flop forward scan, one thread per sequence n.
// Computes logZ[n] = log(8) + sum_t factors_t
// ---------------------------------------------------------------------------
__global__ void scan_kernel(const float* __restrict__ y, float* __restrict__ logZ)
{
    int n = blockIdx.x * blockDim.x + threadIdx.x;
    if (n >= NSEQ) return;

    const float LOG8 = 2.0794415416798357f;
    float fwd[8];
    #pragma unroll
    for (int i = 0; i < 8; ++i) fwd[i] = -LOG8;
    float lz = LOG8;

    for (int t = 0; t < TSEQ; ++t) {
        const float* sc = y + ((size_t)t * NSEQ + n) * CDIM;
        float s[CDIM];
        const v4f* scv = (const v4f*)sc;   // 160B stride -> 16B aligned
        #pragma unroll
        for (int q = 0; q < CDIM / 4; ++q) {
            v4f v = scv[q];
            s[q * 4 + 0] = v.x; s[q * 4 + 1] = v.y;
            s[q * 4 + 2] = v.z; s[q * 4 + 3] = v.w;
        }

        float ns[8];
        // flip destinations: logsumexp over all 8 source states
        #pragma unroll
        for (int j = 0; j < 4; ++j) {
            float vv[8], m = -INFINITY;
            #pragma unroll
            for (int i = 0; i < 8; ++i) {
                vv[i] = fwd[i] + s[j * 8 + i];
                m = fmaxf(m, vv[i]);
            }
            float sum = 0.0f;
            #pragma unroll
            for (int i = 0; i < 8; ++i) sum += __expf(vv[i] - m);
            ns[j] = m + __logf(sum);
        }
        // flop destinations: logaddexp of 2 sources
        #pragma unroll
        for (int j = 0; j < 4; ++j) {
            float a = fwd[j]     + s[32 + j];
            float c = fwd[4 + j] + s[36 + j];
            float m = fmaxf(a, c);
            ns[4 + j] = m + __logf(__expf(a - m) + __expf(c - m));
        }
        // normalization factor
        float m = -INFINITY;
        #pragma unroll
        for (int i = 0; i < 8; ++i) m = fmaxf(m, ns[i]);
        float sum = 0.0f;
        #pragma unroll
        for (int i = 0; i < 8; ++i) sum += __expf(ns[i] - m);
        float f = m + __logf(sum);
        #pragma unroll
        for (int i = 0; i < 8; ++i) fwd[i] = ns[i] - f;
        lz += f;
    }
    logZ[n] = lz;
}

// ---------------------------------------------------------------------------
// Kernel 3: out[t,n,c] -= logZ[n] / T   (in place; kernel 1 rewrites out
// every launch, so this stays deterministic across graph replays)
// ---------------------------------------------------------------------------
__global__ void norm_kernel(float* __restrict__ out,
                            const float* __restrict__ logZ, size_t total)
{
    size_t i = (size_t)blockIdx.x * blockDim.x + threadIdx.x;
    if (i >= total) return;
    int n = (int)((i / CDIM) % NSEQ);
    out[i] -= logZ[n] * (1.0f / (float)TSEQ);
}

// ---------------------------------------------------------------------------
extern "C" void kernel_launch(void* const* d_in, const int* in_sizes, int n_in,
                              void* d_out, int out_size, void* d_ws, size_t ws_size,
                              hipStream_t stream)
{
    const float* x  = (const float*)d_in[0];   // (2048, 256, 256)
    const float* W  = (const float*)d_in[1];   // (40, 256)
    const float* b  = (const float*)d_in[2];   // (40,)
    float* out  = (float*)d_out;               // (2048, 256, 40)
    float* logZ = (float*)d_ws;                // 256 floats scratch

    const int M = in_sizes[0] / KDIM;          // 524288 rows
    const int gemm_blocks = M / MTILE;         // 4096

    gemm_tanh_kernel<<<gemm_blocks, 256, 0, stream>>>(x, W, b, out);
    scan_kernel<<<NSEQ / 32, 32, 0, stream>>>(out, logZ);

    size_t total = (size_t)M * CDIM;
    norm_kernel<<<(unsigned)((total + 255) / 256), 256, 0, stream>>>(out, logZ, total);
}